// GCN_28561532519039
// MI455X (gfx1250) — compile-verified
//
#include <hip/hip_runtime.h>
#include <stdint.h>

typedef float v2f __attribute__((ext_vector_type(2)));
typedef float v8f __attribute__((ext_vector_type(8)));

#define N_   128
#define C_   64
#define T_   128
#define V_   25
#define H_   3
#define HC_  192   // H*C
#define TB   2     // t-values per workgroup
#define COLS 64    // TB * 32 (V padded to 32)
#define KP   28    // u padded to multiple of 4

// LDS layout (float offsets). K dimensions are pair-interleaved so every WMMA
// operand is one contiguous ds_load_b64:
//   At2[h][u>>1][v][u&1], Wl2[hc>>1][o][hc&1], Y2[hc>>1][col][hc&1]
#define XS_OFF 0
#define XS_SZ  (C_ * COLS)           // 4096  : Xs[c][tt*32+u], u padded
#define AT_OFF (XS_OFF + XS_SZ)      // 4096
#define AT_SZ  (H_ * (KP/2) * 32 * 2)// 2688
#define WL_OFF (AT_OFF + AT_SZ)      // 6784
#define WL_SZ  ((HC_/2) * 64 * 2)    // 12288
#define Y_OFF  (WL_OFF + WL_SZ)      // 19072
#define Y_SZ   ((HC_/2) * COLS * 2)  // 12288
#define BN_OFF (Y_OFF + Y_SZ)        // 31360 : scale[64], shift[64]
#define LDS_FLOATS (BN_OFF + 128)    // 31488 floats = 125952 bytes

// Async global->LDS scatter (GVS mode: uniform 64-bit SGPR base + per-lane
// 32-bit byte offset; LDS dest byte address in a VGPR). Tracked by ASYNCcnt.
__device__ __forceinline__ void async_g2l_b32(void* lds_ptr, const void* base,
                                              uint32_t byte_off) {
    uint32_t lws = (uint32_t)(uintptr_t)lds_ptr;   // flat LDS addr[31:0] == LDS offset
    asm volatile("global_load_async_to_lds_b32 %0, %1, %2"
                 :: "v"(lws), "v"(byte_off), "s"(base) : "memory");
}

__device__ __forceinline__ void wait_async0() {
#if __has_builtin(__builtin_amdgcn_s_wait_asynccnt)
    __builtin_amdgcn_s_wait_asynccnt(0);
#else
    asm volatile("s_wait_asynccnt 0x0" ::: "memory");
#endif
}

#define WMMA_F32(Aop, Bop, Cacc) \
    __builtin_amdgcn_wmma_f32_16x16x4_f32(false, (Aop), false, (Bop), (short)0, (Cacc), false, false)

__device__ __forceinline__ v2f lds_b64(const float* p) {
    return *(const v2f*)p;
}

__global__ __launch_bounds__(256)
void gcn_fused(const float* __restrict__ x,   const float* __restrict__ A,
               const float* __restrict__ Wd,  const float* __restrict__ bd,
               const float* __restrict__ gam, const float* __restrict__ bet,
               const float* __restrict__ rmean, const float* __restrict__ rvar,
               float* __restrict__ out)
{
    extern __shared__ float lds[];
    const int tid  = threadIdx.x;
    const int wave = tid >> 5;
    const int lane = tid & 31;
    const int m16  = lane & 15;
    const int hi   = lane >> 4;      // lane half: holds K+2/K+3 of A/B operands
    const int ksel = hi * 2;
    const int n      = blockIdx.y;
    const int tblock = blockIdx.x;   // 0..T/TB-1
    const int t0     = tblock * TB;

    // ---- stage Xs[c][tt*32+u] via async global->LDS; pads zeroed via ds ----
    for (int s = tid; s < XS_SZ; s += 256) {
        int c = s >> 6, rem = s & 63, tt = rem >> 5, u = rem & 31;
        if (u < V_) {
            uint32_t goff = (uint32_t)((((n * C_ + c) * T_ + (t0 + tt)) * V_ + u) * 4);
            async_g2l_b32(&lds[XS_OFF + s], x, goff);
        } else {
            lds[XS_OFF + s] = 0.f;
        }
    }
    // ---- stage At2[h][u>>1][v][u&1] = A[h][v][u] (async), zero-padded ----
    for (int s = tid; s < AT_SZ; s += 256) {
        int h = s / ((KP/2)*64), rem = s % ((KP/2)*64);
        int kp = rem >> 6, rem2 = rem & 63, v = rem2 >> 1, e = rem2 & 1;
        int u = kp * 2 + e;
        if (u < V_ && v < V_) {
            uint32_t goff = (uint32_t)(((h * V_ + v) * V_ + u) * 4);
            async_g2l_b32(&lds[AT_OFF + s], A, goff);
        } else {
            lds[AT_OFF + s] = 0.f;
        }
    }
    // ---- stage Wl2[hc>>1][o][hc&1] = Wd[h][o][c] : transpose-gather ----
    for (int s = tid; s < WL_SZ; s += 256) {
        int hcp = s >> 7, rem = s & 127, o = rem >> 1, e = rem & 1;
        int hc = hcp * 2 + e, h = hc >> 6, c = hc & 63;
        uint32_t goff = (uint32_t)(((h * C_ + o) * C_ + c) * 4);
        async_g2l_b32(&lds[WL_OFF + s], Wd, goff);
    }
    // ---- fold bias-sum + BN into scale/shift ----
    if (tid < C_) {
        int o = tid;
        float bsum = bd[o] + bd[C_ + o] + bd[2 * C_ + o];
        float inv  = gam[o] * rsqrtf(rvar[o] + 1e-5f);
        lds[BN_OFF + o]      = inv;
        lds[BN_OFF + 64 + o] = (bsum - rmean[o]) * inv + bet[o];
    }
    wait_async0();        // ASYNCcnt not covered by __syncthreads' waits
    __syncthreads();

    // ================= Phase A =================
    // Y[h*64+c][col] = sum_u A[h,v,u] * x[n,c,t,u]
    // WMMA: M=c(16), N=v(16), K=u(28, 7 steps).
    // 24 pairs (h,ctile,tt), 3 per wave; two v-halves share the x A-operand.
    for (int i = 0; i < 3; ++i) {
        int p = wave * 3 + i;
        int h = p >> 3, rem = p & 7, ctile = rem >> 1, tt = rem & 1;
        int cbase = ctile * 16;
        const float* Xrow = &lds[XS_OFF + (cbase + m16) * COLS + tt * 32 + ksel];
        // At2: row-pair index kp = k0/2 + hi  -> fold hi into base
        const float* Ar   = &lds[AT_OFF + h * ((KP/2)*64) + hi * 64 + m16 * 2];
        v8f acc0 = {}, acc1 = {};
#pragma unroll
        for (int k0 = 0; k0 < KP; k0 += 4) {
            v2f a  = lds_b64(Xrow + k0);             // x rows, contiguous K pair
            v2f b0 = lds_b64(Ar + (k0/2) * 64);      // v 0..15
            v2f b1 = lds_b64(Ar + (k0/2) * 64 + 32); // v 16..31
            acc0 = WMMA_F32(a, b0, acc0);
            acc1 = WMMA_F32(a, b1, acc1);
        }
        // scatter C-layout tiles into Y2 as row pairs (b64 stores):
        // rows hcb+2r (e=0) and hcb+2r+1 (e=1) are adjacent in Y2.
        int colw = tt * 32 + m16;
        int hcb  = h * C_ + cbase + hi * 8;          // even
#pragma unroll
        for (int r = 0; r < 4; ++r) {
            int hcp = (hcb >> 1) + r;
            v2f p0 = { acc0[2*r], acc0[2*r + 1] };
            v2f p1 = { acc1[2*r], acc1[2*r + 1] };
            *(v2f*)&lds[Y_OFF + hcp * (COLS*2) + colw * 2]        = p0;
            *(v2f*)&lds[Y_OFF + hcp * (COLS*2) + (colw + 16) * 2] = p1;
        }
    }
    __syncthreads();

    // ================= Phase B =================
    // out[o][col] = sum_hc Wl[hc][o] * Y[hc][col]
    // WMMA: M=o(16), N=col(16), K=hc(192, 48 steps).
    // 8 waves = 4 otiles x 2 col-pairs; 2 tiles share the W A-operand.
    {
        int otile = wave >> 1, jpair = wave & 1;
        int obase = otile * 16, colbase = jpair * 32;
        const float* Wp = &lds[WL_OFF + hi * 128 + (obase + m16) * 2];
        const float* Yp = &lds[Y_OFF + hi * 128 + (colbase + m16) * 2];
        v8f acc0 = {}, acc1 = {};
#pragma unroll 8
        for (int k0 = 0; k0 < HC_; k0 += 4) {
            v2f a  = lds_b64(Wp + (k0/2) * 128);
            v2f b0 = lds_b64(Yp + (k0/2) * 128);
            v2f b1 = lds_b64(Yp + (k0/2) * 128 + 32);
            acc0 = WMMA_F32(a, b0, acc0);
            acc1 = WMMA_F32(a, b1, acc1);
        }
        // epilogue: BN + residual + ReLU, store (v>=25 pad lanes drop out)
        auto epilogue = [&](const v8f& acc, int col) {
            int tt = col >> 5, v = col & 31;
            if (v < V_) {
                int t = t0 + tt;
#pragma unroll
                for (int r = 0; r < 8; ++r) {
                    int o = obase + r + hi * 8;
                    float val = acc[r] * lds[BN_OFF + o] + lds[BN_OFF + 64 + o];
                    val += lds[XS_OFF + o * COLS + tt * 32 + v];  // residual (c==o)
                    val = fmaxf(val, 0.f);
                    out[((n * C_ + o) * T_ + t) * V_ + v] = val;
                }
            }
        };
        epilogue(acc0, colbase + m16);
        epilogue(acc1, colbase + 16 + m16);
    }
}

extern "C" void kernel_launch(void* const* d_in, const int* in_sizes, int n_in,
                              void* d_out, int out_size, void* d_ws, size_t ws_size,
                              hipStream_t stream) {
    const float* x     = (const float*)d_in[0];
    const float* A     = (const float*)d_in[1];
    const float* Wd    = (const float*)d_in[2];
    const float* bd    = (const float*)d_in[3];
    const float* gam   = (const float*)d_in[4];
    const float* bet   = (const float*)d_in[5];
    const float* rmean = (const float*)d_in[6];
    const float* rvar  = (const float*)d_in[7];
    float* out = (float*)d_out;

    dim3 grid(T_ / TB, N_);   // 64 x 128 workgroups
    dim3 block(256);          // 8 waves (wave32)
    size_t shmem = LDS_FLOATS * sizeof(float);   // ~123 KB; 2 WGs fit in 320 KB WGP LDS
    gcn_fused<<<grid, block, shmem, stream>>>(x, A, Wd, bd, gam, bet, rmean, rvar, out);
}